// DeepBayesianFilterBlockDiag_62629213110532
// MI455X (gfx1250) — compile-verified
//
#include <hip/hip_runtime.h>
#include <hip/hip_bf16.h>

// ---------------- problem constants (from reference) ----------------
#define Bb      4
#define Tt      64
#define Zz      32
#define LATENT  64              // Z * DIM
#define MROWS   256             // B * T
#define D_OBS   152064          // H*W + 4*L*H*W
#define NTILES  (D_OBS / 16)    // 9504 column tiles of 16
#define GRID_B  304             // workgroups for the fused GEMM+NLL kernel

typedef float v2f __attribute__((ext_vector_type(2)));
typedef float v8f __attribute__((ext_vector_type(8)));

// ---------------- workspace layout ----------------
// [0,               65536)  : X latent matrix, f32 [256 x 64] row-major
// [65536,           65664)  : KL per-block partials (32 floats)
// [65664,           66880)  : NLL per-block partials (GRID_B floats)
#define WS_X_OFF    0
#define WS_KL_OFF   65536
#define WS_NLL_OFF  65664

// =====================================================================
// Kernel A: per-2x2-block KL + Cholesky sample -> X, KL partial sums.
// 8192 blocks total, 1 thread each. grid = 32 x 256.
// =====================================================================
__global__ void kl_sample_kernel(const float* __restrict__ mu_f,
                                 const float* __restrict__ sig_f,
                                 const float* __restrict__ mu_p,
                                 const float* __restrict__ sig_p,
                                 const float* __restrict__ eps,
                                 float* __restrict__ Xout,
                                 float* __restrict__ klPart) {
    __shared__ float red[256];
    const int tid = threadIdx.x;
    const int idx = blockIdx.x * 256 + tid;      // (b*T+t)*Z + z

    // sigma_pred block (row-major 2x2)
    const float a  = sig_p[idx * 4 + 0];
    const float b  = sig_p[idx * 4 + 1];
    const float c  = sig_p[idx * 4 + 2];
    const float d  = sig_p[idx * 4 + 3];
    // sigma_filtered block
    const float af = sig_f[idx * 4 + 0];
    const float bf = sig_f[idx * 4 + 1];
    const float cf = sig_f[idx * 4 + 2];
    const float df = sig_f[idx * 4 + 3];

    const float det_q = a * d - b * c;
    const float det_p = af * df - bf * cf;
    const float inv_det_q = 1.0f / det_q;

    // trace(inv_q @ sig_f) via elementwise form of [[d,-b],[-c,a]]/det
    const float trace_term = (d * af - b * bf - c * cf + a * df) * inv_det_q;

    const float m0 = mu_f[idx * 2 + 0];
    const float m1 = mu_f[idx * 2 + 1];
    const float dx = mu_p[idx * 2 + 0] - m0;
    const float dy = mu_p[idx * 2 + 1] - m1;
    const float quad_term = (d * dx * dx - (b + c) * dx * dy + a * dy * dy) * inv_det_q;

    const float kl = 0.5f * (trace_term + quad_term - 2.0f + __logf(det_q) - __logf(det_p));

    // closed-form Cholesky of sig_f; sample x = mu + L @ eps
    const float l11 = __fsqrt_rn(af);
    const float l21 = cf / l11;
    const float l22 = __fsqrt_rn(df - l21 * l21);
    const float e0 = eps[idx * 2 + 0];
    const float e1 = eps[idx * 2 + 1];
    Xout[idx * 2 + 0] = m0 + l11 * e0;
    Xout[idx * 2 + 1] = m1 + l21 * e0 + l22 * e1;

    // deterministic tree reduce of KL within the block
    red[tid] = kl;
    __syncthreads();
    for (int s = 128; s > 0; s >>= 1) {
        if (tid < s) red[tid] += red[tid + s];
        __syncthreads();
    }
    if (tid == 0) klPart[blockIdx.x] = red[0];
}

// =====================================================================
// Kernel B: fused  (X @ W_dec + b_dec)  ->  Gaussian NLL  reduction.
// One wave32 owns a 16-wide column tile; C tile never leaves VGPRs.
// WMMA: V_WMMA_F32_16X16X4_F32, K=64 -> 16 k-steps; 16 m-tiles of 16.
// =====================================================================
__global__ void gemm_nll_kernel(const float* __restrict__ X,
                                const float* __restrict__ W,
                                const float* __restrict__ bdec,
                                const float* __restrict__ logR,
                                const float* __restrict__ target,
                                float* __restrict__ nllPart) {
    extern __shared__ float smem[];          // 64KB X stage + 1KB reduce
    float* ldsX = smem;                      // [256][64]
    float* red  = smem + MROWS * LATENT;     // [256]

    const int tid = threadIdx.x;

    // stage the whole latent matrix X (64 KB) into LDS once per workgroup
    for (int i = tid; i < MROWS * LATENT; i += 256) ldsX[i] = X[i];
    __syncthreads();

    const int lane   = tid & 31;
    const int laneN  = lane & 15;            // N column within tile / M row for A
    const int laneHi = lane >> 4;            // selects K pair (A/B) or M+8 (C)
    const int gwave  = blockIdx.x * 8 + (tid >> 5);
    const int nWaves = GRID_B * 8;

    const float LOG2PI = 1.8378770664093453f;
    float partial = 0.0f;

    for (int nt = gwave; nt < NTILES; nt += nWaves) {
        const int col = nt * 16 + laneN;

        // resident B fragments for this column tile: 16 k-steps x v2f.
        // B 4x16 layout: lane = N; lanes 0-15 carry K rows {0,1} in the two
        // VGPRs, lanes 16-31 carry K rows {2,3}.
        v2f bfr[16];
#pragma unroll
        for (int k = 0; k < 16; ++k) {
            const int k0 = 4 * k + 2 * laneHi;
            bfr[k].x = W[(size_t)k0 * D_OBS + col];
            bfr[k].y = W[(size_t)(k0 + 1) * D_OBS + col];
        }

        // per-column epilogue constants (one lane <-> one column)
        const float bd    = bdec[col];
        const float lr    = logR[col];
        const float invR  = __expf(-2.0f * lr);
        const float cterm = 0.5f * (LOG2PI + 2.0f * lr);

        for (int mt = 0; mt < 16; ++mt) {
            v8f acc = {};                    // C starts at 0; b_dec folded later
            const int arow = mt * 16 + laneN;    // A: lane = M
#pragma unroll
            for (int k = 0; k < 16; ++k) {
                const int k0 = 4 * k + 2 * laneHi;   // A: lane-half selects K pair
                const float2 av =
                    *(const float2*)&ldsX[arow * LATENT + k0];   // ds_load_b64
                v2f afr; afr.x = av.x; afr.y = av.y;
                acc = __builtin_amdgcn_wmma_f32_16x16x4_f32(
                    false, afr, false, bfr[k], (short)0, acc, false, false);
            }
            // epilogue: C VGPR r -> row mt*16 + r + 8*laneHi, col = lane%16
#pragma unroll
            for (int r = 0; r < 8; ++r) {
                const int row = mt * 16 + r + 8 * laneHi;
                const float tgt = target[(size_t)row * D_OBS + col];
                const float diff = tgt - (acc[r] + bd);
                partial += cterm + 0.5f * diff * diff * invR;
            }
        }
    }

    // deterministic workgroup reduction
    red[tid] = partial;
    __syncthreads();
    for (int s = 128; s > 0; s >>= 1) {
        if (tid < s) red[tid] += red[tid + s];
        __syncthreads();
    }
    if (tid == 0) nllPart[blockIdx.x] = red[0];
}

// =====================================================================
// Kernel C: final deterministic scalar combine  (nll + kl) / B
// =====================================================================
__global__ void finalize_kernel(const float* __restrict__ klPart,
                                const float* __restrict__ nllPart,
                                float* __restrict__ out) {
    if (threadIdx.x == 0 && blockIdx.x == 0) {
        float kl = 0.0f;
        for (int i = 0; i < 32; ++i) kl += klPart[i];
        float nll = 0.0f;
        for (int i = 0; i < GRID_B; ++i) nll += nllPart[i];
        out[0] = (nll + kl) * (1.0f / (float)Bb);
    }
}

extern "C" void kernel_launch(void* const* d_in, const int* in_sizes, int n_in,
                              void* d_out, int out_size, void* d_ws, size_t ws_size,
                              hipStream_t stream) {
    const float* mu_f  = (const float*)d_in[0];
    const float* sig_f = (const float*)d_in[1];
    const float* mu_p  = (const float*)d_in[2];
    const float* sig_p = (const float*)d_in[3];
    const float* tgt   = (const float*)d_in[4];
    const float* W     = (const float*)d_in[5];
    const float* bdec  = (const float*)d_in[6];
    const float* logR  = (const float*)d_in[7];
    const float* eps   = (const float*)d_in[8];

    char* ws = (char*)d_ws;
    float* Xbuf    = (float*)(ws + WS_X_OFF);
    float* klPart  = (float*)(ws + WS_KL_OFF);
    float* nllPart = (float*)(ws + WS_NLL_OFF);
    float* out     = (float*)d_out;

    // A: KL + reparameterized sample (8192 2x2 blocks)
    kl_sample_kernel<<<32, 256, 0, stream>>>(mu_f, sig_f, mu_p, sig_p, eps,
                                             Xbuf, klPart);

    // B: fused WMMA GEMM + Gaussian NLL (dynamic LDS: 64KB X + 1KB reduce)
    const size_t smemB = (MROWS * LATENT + 256) * sizeof(float);
    gemm_nll_kernel<<<GRID_B, 256, smemB, stream>>>(Xbuf, W, bdec, logR, tgt,
                                                    nllPart);

    // C: deterministic scalar combine
    finalize_kernel<<<1, 32, 0, stream>>>(klPart, nllPart, out);
}